// GraphAttention_70111046140613
// MI455X (gfx1250) — compile-verified
//
#include <hip/hip_runtime.h>
#include <math.h>

#define N_NODES 4096
#define F_IN    512
#define F_OUT   256
#define ALPHA   0.2f

typedef float v2f __attribute__((ext_vector_type(2)));
typedef float v8f __attribute__((ext_vector_type(8)));
typedef int   v4i __attribute__((ext_vector_type(4)));

#if __has_builtin(__builtin_amdgcn_global_load_async_to_lds_b128) && \
    __has_builtin(__builtin_amdgcn_s_wait_asynccnt)
#define USE_ASYNC_LDS 1
#else
#define USE_ASYNC_LDS 0
#endif

// AS-qualified vector pointer types for the async builtin:
//   global (HIP __device__, target AS1) source, LDS (__shared__, AS3) dest.
typedef __attribute__((address_space(1))) v4i gbl_v4i;
typedef __attribute__((address_space(3))) v4i lds_v4i;

// Copy one 16-byte segment global -> LDS (async ASYNCcnt path when available).
__device__ __forceinline__ void copy16_g2l(float* lds_dst, const float* g_src) {
#if USE_ASYNC_LDS
  __builtin_amdgcn_global_load_async_to_lds_b128(
      (gbl_v4i*)g_src, (lds_v4i*)lds_dst, 0, 0);
#else
  *(float4*)lds_dst = *(const float4*)g_src;
#endif
}

__device__ __forceinline__ void wait_lds_copies() {
#if USE_ASYNC_LDS
  __builtin_amdgcn_s_wait_asynccnt(0);
#endif
  __syncthreads();
}

// ---------------------------------------------------------------------------
// fp32 WMMA GEMM: C[M x Nc] = A[M x K] * B[K x Nc], row-major fp32.
// Block = 256 threads (8 waves) -> one 16(M) x 128(N) C tile; wave w owns the
// 16x16 sub-tile at column w*16. K is stepped in chunks of 32 staged in LDS:
//   Alds[16][32] (2 KB)  : coalesced async copy, read once per block
//   Blds[32][128] (16 KB): coalesced async copy (512B-contig rows block-wide)
// Fragment layouts per CDNA5 ISA 7.12.2 (V_WMMA_F32_16X16X4_F32):
//   A 16x4 : lanes 0-15 {K=k,k+1}, lanes 16-31 {K=k+2,k+3}, row = lane&15
//   B 4x16 : VGPR0 rows k / k+2, VGPR1 rows k+1 / k+3, col = lane&15
//   C 16x16: VGPR v -> row v (lanes 0-15) / v+8 (lanes 16-31), col = lane&15
// Requires: K % 32 == 0, Nc % 128 == 0.
// ---------------------------------------------------------------------------
__global__ void __launch_bounds__(256)
wmma_gemm_f32(const float* __restrict__ A, const float* __restrict__ B,
              float* __restrict__ C, int M, int Nc, int K) {
  __shared__ float Alds[16 * 32];
  __shared__ float Blds[32 * 128];

  const int t    = threadIdx.x;
  const int lane = t & 31;
  const int w    = t >> 5;        // wave id 0..7
  const int half = lane >> 4;     // 0: K pair {0,1}, 1: K pair {2,3}
  const int lr   = lane & 15;
  const int tm   = blockIdx.x * 16;          // M tile origin
  const int tnb  = blockIdx.y * 128;         // N group origin (block)
  const int wn   = w * 16;                   // wave's column offset in group

  // Static copy assignments (16B segments):
  //   A slab: 128 segs -> threads 0..127, row = t>>3, seg = t&7
  //   B slab: 1024 segs -> 4 per thread, seg = t + it*256, row = seg>>5
  const int ar = t >> 3, as = (t & 7) * 4;

  v8f acc = {};
  for (int k0 = 0; k0 < K; k0 += 32) {
    if (t < 128)
      copy16_g2l(&Alds[ar * 32 + as], A + (size_t)(tm + ar) * K + k0 + as);
#pragma unroll
    for (int it = 0; it < 4; ++it) {
      const int seg = t + it * 256;
      const int kr = seg >> 5, sc = (seg & 31) * 4;
      copy16_g2l(&Blds[kr * 128 + sc], B + (size_t)(k0 + kr) * Nc + tnb + sc);
    }
    if (t < 128 && k0 + 32 < K)   // stride-ahead HBM prefetch of next A slab
      __builtin_prefetch(A + (size_t)(tm + ar) * K + k0 + 32 + as, 0, 1);

    wait_lds_copies();

#pragma unroll
    for (int kk = 0; kk < 32; kk += 4) {
      const int ka = kk + 2 * half;
      v2f afrag = *(const v2f*)&Alds[lr * 32 + ka];
      v2f bfrag;
      bfrag.x = Blds[(ka)     * 128 + wn + lr];
      bfrag.y = Blds[(ka + 1) * 128 + wn + lr];
      acc = __builtin_amdgcn_wmma_f32_16x16x4_f32(
          /*neg_a=*/false, afrag, /*neg_b=*/false, bfrag,
          /*c_mod=*/(short)0, acc, /*reuse_a=*/false, /*reuse_b=*/false);
    }
    __syncthreads();   // protect LDS before next chunk overwrites
  }

  const int tn = tnb + wn;
#pragma unroll
  for (int v = 0; v < 8; ++v)
    C[(size_t)(tm + v + 8 * half) * Nc + tn + lr] = acc[v];
}

// ---------------------------------------------------------------------------
// Fused per-row pass: degree -> keep flag, plus s1 = Wh.a1, s2 = Wh.a2.
// One 256-thread block per row. Exact compare (sums of 1.0s are exact fp32).
// ---------------------------------------------------------------------------
__global__ void __launch_bounds__(256)
row_stats_kernel(const float* __restrict__ adj, const float* __restrict__ Wh,
                 const float* __restrict__ a,
                 float* __restrict__ s1, float* __restrict__ s2,
                 float* __restrict__ keep) {
  __shared__ float sd[256], sa[256], sb[256];
  const int i = blockIdx.x;
  const int t = threadIdx.x;

  float deg = 0.0f;
  const float4* arow = (const float4*)(adj + (size_t)i * N_NODES);
  for (int j = t; j < N_NODES / 4; j += 256) {
    float4 v = arow[j];
    deg += v.x + v.y + v.z + v.w;
  }
  const float p = Wh[(size_t)i * F_OUT + t];   // F_OUT == blockDim.x == 256
  sd[t] = deg;
  sa[t] = p * a[t];
  sb[t] = p * a[F_OUT + t];
  __syncthreads();
  for (int s = 128; s > 0; s >>= 1) {
    if (t < s) { sd[t] += sd[t + s]; sa[t] += sa[t + s]; sb[t] += sb[t + s]; }
    __syncthreads();
  }
  if (t == 0) {
    keep[i] = (sd[0] != 1.0f) ? 1.0f : 0.0f;
    s1[i]   = sa[0];
    s2[i]   = sb[0];
  }
}

// ---------------------------------------------------------------------------
// Masked online-softmax row statistics: rowmax / rowsum over
// e[i,j] = leaky_relu(s1[i] + s2[j]) where adj[i,j] > 0 and keep[j].
// ---------------------------------------------------------------------------
__global__ void __launch_bounds__(256)
softmax_stats_kernel(const float* __restrict__ adj, const float* __restrict__ s1,
                     const float* __restrict__ s2, const float* __restrict__ keep,
                     float* __restrict__ rowmax, float* __restrict__ rowsum) {
  __shared__ float sm[256], ss[256];
  const int i = blockIdx.x;
  const int t = threadIdx.x;

  if (keep[i] == 0.0f) {
    if (t == 0) { rowmax[i] = 0.0f; rowsum[i] = 1.0f; }  // dummies, unused
    return;
  }
  const float s1i = s1[i];
  const float* arow = adj + (size_t)i * N_NODES;

  float m = -INFINITY, s = 0.0f;
  for (int j = t; j < N_NODES; j += 256) {
    if (arow[j] > 0.0f && keep[j] != 0.0f) {
      float v = s1i + s2[j];
      v = (v > 0.0f) ? v : ALPHA * v;
      if (v > m) { s = s * __expf(m - v) + 1.0f; m = v; }
      else       { s += __expf(v - m); }
    }
  }
  sm[t] = m; ss[t] = s;
  __syncthreads();
  for (int st = 128; st > 0; st >>= 1) {
    if (t < st) {
      const float m2 = sm[t + st], sB = ss[t + st];
      const float mm = fmaxf(sm[t], m2);
      ss[t] = (mm == -INFINITY) ? 0.0f
            : ss[t] * __expf(sm[t] - mm) + sB * __expf(m2 - mm);
      sm[t] = mm;
    }
    __syncthreads();
  }
  if (t == 0) { rowmax[i] = sm[0]; rowsum[i] = ss[0]; }
}

// ---------------------------------------------------------------------------
// Write final_attn = attn + diag(~keep). float4 coalesced over columns.
// h_prime = final_attn @ Wh then reproduces where(keep, attn@Wh, Wh) exactly,
// since !keep rows are one-hot at the diagonal.
// ---------------------------------------------------------------------------
__global__ void __launch_bounds__(256)
attn_write_kernel(const float* __restrict__ adj, const float* __restrict__ s1,
                  const float* __restrict__ s2, const float* __restrict__ keep,
                  const float* __restrict__ rowmax, const float* __restrict__ rowsum,
                  float* __restrict__ final_attn) {
  const int i  = blockIdx.y;
  const int j0 = (blockIdx.x * 256 + threadIdx.x) * 4;

  const float ki  = keep[i];
  const float s1i = s1[i];
  const float m   = rowmax[i];
  const float inv = 1.0f / rowsum[i];

  float4 av = *(const float4*)(adj + (size_t)i * N_NODES + j0);
  const float* ap = &av.x;
  float4 out;
  float* o = &out.x;
#pragma unroll
  for (int c = 0; c < 4; ++c) {
    const int j = j0 + c;
    float at = 0.0f;
    if (ki != 0.0f && ap[c] > 0.0f && keep[j] != 0.0f) {
      float v = s1i + s2[j];
      v = (v > 0.0f) ? v : ALPHA * v;
      at = __expf(v - m) * inv;
    }
    if (j == i && ki == 0.0f) at = 1.0f;   // + diag(~keep)
    o[c] = at;
  }
  *(float4*)(final_attn + (size_t)i * N_NODES + j0) = out;
}

// ---------------------------------------------------------------------------
// Inputs (setup_inputs order): h (4096x512 f32), adj (4096x4096 f32),
// W (512x256 f32), a (512 f32). Output: h_prime (4096x256) ++ final_attn
// (4096x4096), fp32, flat.
// ---------------------------------------------------------------------------
extern "C" void kernel_launch(void* const* d_in, const int* in_sizes, int n_in,
                              void* d_out, int out_size, void* d_ws, size_t ws_size,
                              hipStream_t stream) {
  (void)in_sizes; (void)n_in; (void)out_size; (void)ws_size;

  const float* h   = (const float*)d_in[0];
  const float* adj = (const float*)d_in[1];
  const float* W   = (const float*)d_in[2];
  const float* a   = (const float*)d_in[3];

  float* out        = (float*)d_out;
  float* h_prime    = out;                                 // 4096*256
  float* final_attn = out + (size_t)N_NODES * F_OUT;       // 4096*4096

  char*  ws     = (char*)d_ws;
  float* Wh     = (float*)ws;                                        // 4 MB
  float* s1     = (float*)(ws + (size_t)N_NODES * F_OUT * sizeof(float));
  float* s2     = s1 + N_NODES;
  float* keep   = s2 + N_NODES;
  float* rowmax = keep + N_NODES;
  float* rowsum = rowmax + N_NODES;

  // 1) Wh = h @ W  (fp32 WMMA, async-LDS staged)
  wmma_gemm_f32<<<dim3(N_NODES / 16, F_OUT / 128), 256, 0, stream>>>(
      h, W, Wh, N_NODES, F_OUT, F_IN);

  // 2) degree/keep + attention projections s1, s2
  row_stats_kernel<<<N_NODES, 256, 0, stream>>>(adj, Wh, a, s1, s2, keep);

  // 3) masked softmax row stats
  softmax_stats_kernel<<<N_NODES, 256, 0, stream>>>(adj, s1, s2, keep, rowmax, rowsum);

  // 4) final_attn (second output)
  attn_write_kernel<<<dim3(N_NODES / 1024, N_NODES), 256, 0, stream>>>(
      adj, s1, s2, keep, rowmax, rowsum, final_attn);

  // 5) h_prime = final_attn @ Wh  (fp32 WMMA, handles keep/!keep rows uniformly)
  wmma_gemm_f32<<<dim3(N_NODES / 16, F_OUT / 128), 256, 0, stream>>>(
      final_attn, Wh, h_prime, N_NODES, F_OUT, N_NODES);
}